// ExampleModel_1159641170495
// MI455X (gfx1250) — compile-verified
//
#include <hip/hip_runtime.h>

typedef float v2f __attribute__((ext_vector_type(2)));
typedef float v8f __attribute__((ext_vector_type(8)));

#define THREADS 256
#define MAX_BLOCKS 4096

// items are integer-valued floats in [0, 1e6): int cast is exact.
// keep value if even, else 0.
__device__ __forceinline__ float maskEven(float c) {
    int v = (int)c;
    return (v & 1) ? 0.0f : c;
}

// Stage 1: grid-stride streaming reduction. Each wave accumulates masked
// values into a 16x16 f32 WMMA accumulator via V_WMMA_F32_16X16X4_F32 with a
// B matrix that is 1.0 only in column N=0 (so D[i][0] = rowsum(A_i)+C[i][0],
// all other columns are pass-through). One float per block -> partials[].
__global__ void __launch_bounds__(THREADS) evensum_stage1(
    const float4* __restrict__ in, float* __restrict__ partials, long long nvec)
{
    const int lane = threadIdx.x & 31;

    // B (4x16 f32, 2 VGPRs): rows striped across lanes, lanes encode N.
    // Column N=0 <=> (lane & 15) == 0. Both B VGPRs get 1.0 there.
    v2f b;
    b.x = ((lane & 15) == 0) ? 1.0f : 0.0f;
    b.y = b.x;

    v8f c = {};       // 16x16 f32 WMMA accumulator (8 VGPRs)
    float t = 0.0f;   // scalar accumulator for the divergent remainder

    const long long stride = (long long)gridDim.x * blockDim.x;
    const long long base   = (long long)blockIdx.x * blockDim.x + threadIdx.x;
    const long long nFull  = nvec / stride;   // uniform trip count: EXEC all-1s

    for (long long k = 0; k < nFull; ++k) {
        float4 v = in[base + k * stride];
        v2f a0; a0.x = maskEven(v.x); a0.y = maskEven(v.y);
        v2f a1; a1.x = maskEven(v.z); a1.y = maskEven(v.w);
        // 8 args: (neg_a, A, neg_b, B, c_mod, C, reuse_a, reuse_b)
        c = __builtin_amdgcn_wmma_f32_16x16x4_f32(
                false, a0, false, b, (short)0, c, false, false);
        c = __builtin_amdgcn_wmma_f32_16x16x4_f32(
                false, a1, false, b, (short)0, c, false, false);
    }

    // Remainder (divergent, scalar path only — no WMMA under partial EXEC).
    {
        long long idx = base + nFull * stride;
        if (idx < nvec) {
            float4 v = in[idx];
            t += maskEven(v.x) + maskEven(v.y) + maskEven(v.z) + maskEven(v.w);
        }
    }

    // Column 0 of the accumulator lives in lanes 0 (M=0..7) and 16 (M=8..15).
    float s = c[0] + c[1] + c[2] + c[3] + c[4] + c[5] + c[6] + c[7];
    float contrib = ((lane & 15) == 0) ? s : 0.0f;
    contrib += t;

    // wave32 shuffle reduction
    for (int off = 16; off > 0; off >>= 1)
        contrib += __shfl_down(contrib, off, 32);

    __shared__ float red[THREADS / 32];
    const int wave = threadIdx.x >> 5;
    if (lane == 0) red[wave] = contrib;
    __syncthreads();
    if (wave == 0) {
        float w = (lane < THREADS / 32) ? red[lane] : 0.0f;
        for (int off = 4; off > 0; off >>= 1)
            w += __shfl_down(w, off, 32);
        if (lane == 0) partials[blockIdx.x] = w;
    }
}

// Stage 2: single block, deterministic fixed-order reduction of partials
// (+ any n%4 tail elements handled by thread 0).
__global__ void __launch_bounds__(THREADS) evensum_stage2(
    const float* __restrict__ partials, int nPartials,
    const float* __restrict__ items, long long tailStart, long long n,
    float* __restrict__ out)
{
    float s = 0.0f;
    for (int i = threadIdx.x; i < nPartials; i += THREADS)
        s += partials[i];
    if (threadIdx.x == 0)
        for (long long i = tailStart; i < n; ++i)
            s += maskEven(items[i]);

    const int lane = threadIdx.x & 31;
    const int wave = threadIdx.x >> 5;
    for (int off = 16; off > 0; off >>= 1)
        s += __shfl_down(s, off, 32);

    __shared__ float red[THREADS / 32];
    if (lane == 0) red[wave] = s;
    __syncthreads();
    if (wave == 0) {
        float w = (lane < THREADS / 32) ? red[lane] : 0.0f;
        for (int off = 4; off > 0; off >>= 1)
            w += __shfl_down(w, off, 32);
        if (lane == 0) out[0] = w;
    }
}

extern "C" void kernel_launch(void* const* d_in, const int* in_sizes, int n_in,
                              void* d_out, int out_size, void* d_ws, size_t ws_size,
                              hipStream_t stream) {
    const float* items = (const float*)d_in[0];
    const long long n = (long long)in_sizes[0];
    const long long nvec = n >> 2;          // float4 count
    const long long tailStart = nvec << 2;  // n % 4 leftovers

    long long want = (nvec + THREADS - 1) / THREADS;
    int blocks = (int)(want < 1 ? 1 : (want > MAX_BLOCKS ? MAX_BLOCKS : want));

    float* partials = (float*)d_ws;   // blocks * 4 bytes <= 16 KB scratch

    evensum_stage1<<<blocks, THREADS, 0, stream>>>(
        (const float4*)items, partials, nvec);
    evensum_stage2<<<1, THREADS, 0, stream>>>(
        partials, blocks, items, tailStart, n, (float*)d_out);
}